// CAM_43344809951340
// MI455X (gfx1250) — compile-verified
//
#include <hip/hip_runtime.h>
#include <cstdint>
#include <cstddef>

#define D 32
#define AVW (2 * D)            // av width = 64
#define WPB 8                  // wave32 waves per block = samples per block
#define BLOCK (WPB * 32)

// ---------- CDNA5 feature detection (all guarded; fallbacks always compile) ----
#if defined(__has_builtin)
#  if __has_builtin(__builtin_amdgcn_global_load_async_to_lds_b128)
#    define HAVE_ASYNC_LDS 1
#  endif
#  if __has_builtin(__builtin_amdgcn_tanhf)
#    define FAST_TANH(x) __builtin_amdgcn_tanhf(x)
#  elif __has_builtin(__builtin_amdgcn_tanh_f32)
#    define FAST_TANH(x) __builtin_amdgcn_tanh_f32(x)
#  endif
#endif
#ifndef FAST_TANH
#  define FAST_TANH(x) tanhf(x)     // ocml fallback (software, still compiles)
#endif

#if defined(HAVE_ASYNC_LDS)
// The builtin's params are pointers to 16-byte int vectors in AS1 (global /
// "__device__") and AS3 (LDS / "__shared__") per the round-1 diagnostic.
typedef int v4i_t __attribute__((ext_vector_type(4)));
typedef __attribute__((address_space(1))) v4i_t g_v4i_t;   // global
typedef __attribute__((address_space(3))) v4i_t l_v4i_t;   // LDS
#endif

__global__ __launch_bounds__(BLOCK) void cam_fused_kernel(
    const float* __restrict__ f1, const float* __restrict__ f2,
    const float* __restrict__ Wca, const float* __restrict__ Wcv,
    float* __restrict__ out, int B)
{
    // One av row (64 f32) per wave/sample: sAV[w][0..31]=f1 row, [32..63]=f2 row.
    __shared__ __align__(16) float sAV[WPB][AVW];   // 2 KB

    const int t    = threadIdx.x;
    const int lane = t & 31;
    const int wave = t >> 5;
    const int sampBase = blockIdx.x * WPB;

    // ---- stage this block's 8 av rows into LDS: 128 x 16B chunks --------------
    // chunk c in a row: c<8 -> f1 quad c, c>=8 -> f2 quad (c-8)
    if (t < 128) {
        const int w = t >> 4;
        const int c = t & 15;
        const int s = sampBase + w;
        if (s < B) {
            const float* src = (c < 8) ? (f1 + (size_t)s * D + (c << 2))
                                       : (f2 + (size_t)s * D + ((c - 8) << 2));
            float* dst = &sAV[w][c << 2];
#if defined(HAVE_ASYNC_LDS)
            // CDNA5 async global->LDS DMA: data never touches VGPRs.
            __builtin_amdgcn_global_load_async_to_lds_b128(
                (g_v4i_t*)(uintptr_t)src,
                (l_v4i_t*)(uint32_t)(uintptr_t)dst,
                /*imm offset*/0, /*cpol*/0);
#else
            *(float4*)dst = *(const float4*)src;
#endif
        }
    }
#if defined(HAVE_ASYNC_LDS)
#  if __has_builtin(__builtin_amdgcn_s_wait_asynccnt)
    __builtin_amdgcn_s_wait_asynccnt(0);
#  else
    asm volatile("s_wait_asynccnt 0x0" ::: "memory");
#  endif
#endif
    __syncthreads();

    const int s = sampBase + wave;          // wave-uniform
    if (s >= B) return;

    // Per-lane inputs (lane == output row i).
    const float f1v = sAV[wave][lane];
    const float f2v = sAV[wave][D + lane];
    const float a = f1v * 0.125f;           // SCALE = 1/sqrt(64) = 0.125 exactly
    const float b = f2v * 0.125f;

    float accA = 0.0f, accV = 0.0f;
    // Fully unrolled: av_j is a uniform-address LDS broadcast (DS pipe),
    // W*_j are wave-uniform with constant index -> scalar s_load, consumed as
    // SGPR operands by v_fmac. VALU work per j: 2 mul + 2 v_tanh + 2 fma.
#pragma unroll
    for (int j = 0; j < AVW; ++j) {
        const float avj = sAV[wave][j];
        const float wa  = Wca[j];
        const float wv  = Wcv[j];
        accA = fmaf(FAST_TANH(a * avj), wa, accA);
        accV = fmaf(FAST_TANH(b * avj), wv, accV);
    }

    const float Ha = 0.1f * fmaxf(accA + f1v, 0.0f);
    const float Hv = 0.1f * fmaxf(accV + f2v, 0.0f);

    float* o = out + (size_t)s * AVW;       // [B,1,64] = [H_a | H_v]
    o[lane]     = Ha;
    o[D + lane] = Hv;
}

extern "C" void kernel_launch(void* const* d_in, const int* in_sizes, int n_in,
                              void* d_out, int out_size, void* d_ws, size_t ws_size,
                              hipStream_t stream) {
    const float* f1  = (const float*)d_in[0];   // [B,32]
    const float* f2  = (const float*)d_in[1];   // [B,32]
    const float* Wca = (const float*)d_in[2];   // [1,64]
    const float* Wcv = (const float*)d_in[3];   // [1,64]
    float* out = (float*)d_out;                 // [B,1,64]

    const int B = in_sizes[0] / D;
    const int grid = (B + WPB - 1) / WPB;
    hipLaunchKernelGGL(cam_fused_kernel, dim3(grid), dim3(BLOCK), 0, stream,
                       f1, f2, Wca, Wcv, out, B);
}